// StaticScatterCacheUpdate_39779987096357
// MI455X (gfx1250) — compile-verified
//
#include <hip/hip_runtime.h>
#include <stdint.h>

// ---------------------------------------------------------------------------
// StaticScatterCacheUpdate on MI455X (gfx1250)
//
// B=2, H=32, S=8192, D=128, T=512, f32.
// out = [cache_k with rows position_ids overwritten by new_k,
//        cache_v with rows position_ids overwritten by new_v]  (concat, flat)
//
// Pure HBM streaming problem (~1.06 GiB -> ~46 us @ 23.3 TB/s). Bulk copy
// uses the CDNA5 async global<->LDS DMA path (ASYNCcnt-tracked, no VGPR
// bounce). The IOFFSET immediate of the async ops is added to BOTH the LDS
// and global addresses, so each wave needs only one LDS base VGPR and one
// global base VGPR pair for all 8 slots: 8x async B128 load (offset:0..3584)
// -> s_wait_asynccnt 0 -> 8x async B128 store, all with NT temporal hints
// (both streams exceed the 192 MB L2). The small scatter pass (32 MiB) uses
// plain NT B128 loads/stores.
// ---------------------------------------------------------------------------

typedef float f32x4 __attribute__((ext_vector_type(4)));

constexpr long long Bc = 2, Hc = 32, Sc = 8192, Dc = 128, Tc = 512;
constexpr long long NKV   = Bc * Hc * Sc * Dc;   // 67,108,864 elems per cache
constexpr long long NKV4  = NKV / 4;             // 16,777,216 float4 per cache
constexpr long long NNEW  = Bc * Hc * Tc * Dc;   // 4,194,304 elems per new tensor
constexpr long long NNEW4 = NNEW / 4;            // 1,048,576 float4 per new tensor
constexpr long long DV    = Dc / 4;              // 32 float4 per row

constexpr uint32_t BYTES_PER_TENSOR = (uint32_t)(NKV * 4);    // 256 MiB
constexpr int      WAVES_PER_BLOCK  = 8;                      // 256 threads
constexpr uint32_t WAVE_CHUNK       = 8 * 512;                // 4 KiB per wave
constexpr uint32_t BLOCK_CHUNK      = WAVES_PER_BLOCK * WAVE_CHUNK; // 32 KiB per WG

// Async copy ops: single LDS-base VGPR + single global-base VGPR pair; the
// signed-24-bit IOFFSET advances both sides simultaneously.
#define ASYNC_LOAD(OFF)                                                        \
    asm volatile("global_load_async_to_lds_b128 %0, %1, off offset:" #OFF      \
                 " th:TH_LOAD_NT" :: "v"(lds0), "v"(gsrc) : "memory")
#define ASYNC_STORE(OFF)                                                       \
    asm volatile("global_store_async_from_lds_b128 %0, %1, off offset:" #OFF   \
                 " th:TH_STORE_NT" :: "v"(gdst), "v"(lds0) : "memory")

// ---------------------------------------------------------------------------
// Kernel 1: bulk copy via async LDS staging. blockIdx.y selects k (0) / v (1).
// Each wave owns one contiguous 4 KiB chunk: lane L moves bytes [L*16, L*16+16)
// of each 512 B slot. No loop, no tail (256 MiB / 4 KiB = 65,536 waves).
// ---------------------------------------------------------------------------
__global__ void __launch_bounds__(256)
cache_copy_async_kernel(const void* __restrict__ ck,
                        const void* __restrict__ cv,
                        void* __restrict__ out)
{
    __shared__ alignas(16) unsigned char smem[BLOCK_CHUNK];

    const uint32_t tid  = threadIdx.x;
    const uint32_t wave = tid >> 5;
    const uint32_t lane = tid & 31;

    const uint64_t srcb = (blockIdx.y == 0) ? (uint64_t)ck : (uint64_t)cv;
    const uint64_t dstb = (uint64_t)out + (uint64_t)blockIdx.y * BYTES_PER_TENSOR;

    // Byte offset of this lane's first 16 B within the tensor.
    const uint32_t off0 = (uint32_t)blockIdx.x * BLOCK_CHUNK
                        + wave * WAVE_CHUNK + lane * 16u;
    // LDS byte address (flat-shared low 32 bits == workgroup LDS offset).
    const uint32_t lds0 = (uint32_t)(uintptr_t)(void*)smem
                        + wave * WAVE_CHUNK + lane * 16u;

    const uint64_t gsrc = srcb + off0;
    const uint64_t gdst = dstb + off0;

    // Phase 1: 8 async B128 loads global -> LDS (ASYNCcnt += 8).
    ASYNC_LOAD(0);    ASYNC_LOAD(512);  ASYNC_LOAD(1024); ASYNC_LOAD(1536);
    ASYNC_LOAD(2048); ASYNC_LOAD(2560); ASYNC_LOAD(3072); ASYNC_LOAD(3584);

    // Wait for all loads to land in LDS (loads complete in order).
    asm volatile("s_wait_asynccnt 0x0" ::: "memory");

    // Phase 2: 8 async B128 stores LDS -> global (streaming NT).
    ASYNC_STORE(0);    ASYNC_STORE(512);  ASYNC_STORE(1024); ASYNC_STORE(1536);
    ASYNC_STORE(2048); ASYNC_STORE(2560); ASYNC_STORE(3072); ASYNC_STORE(3584);

    // Outstanding ASYNCcnt drained by s_endpgm's implicit wait-idle.
}

// ---------------------------------------------------------------------------
// Kernel 2: scatter the new rows into the just-written caches.
// One float4 per thread over 2*NNEW4 elements; row slot from position_ids
// (int32 — JAX x64-off demotes int64). Runs after the copy on same stream.
// ---------------------------------------------------------------------------
__global__ void __launch_bounds__(256)
cache_scatter_kernel(const f32x4* __restrict__ nk,
                     const f32x4* __restrict__ nv,
                     const int*   __restrict__ position_ids,
                     f32x4* __restrict__ out)
{
    long long gid = (long long)blockIdx.x * blockDim.x + threadIdx.x;
    if (gid >= 2 * NNEW4) return;

    const bool      is_v = (gid >= NNEW4);
    const long long idx4 = is_v ? (gid - NNEW4) : gid;
    const f32x4*    src  = is_v ? nv : nk;

    const long long dvec = idx4 % DV;
    const long long t    = (idx4 / DV) % Tc;
    const long long bh   = idx4 / (DV * Tc);

    const long long pos  = (long long)position_ids[t];

    const long long half = is_v ? NKV4 : 0;
    const long long dst  = half + (bh * Sc + pos) * DV + dvec;

    f32x4 v = __builtin_nontemporal_load(src + idx4);
    __builtin_nontemporal_store(v, out + dst);
}

// ---------------------------------------------------------------------------
// Host-side launch (graph-capture safe: only kernel launches on `stream`).
// ---------------------------------------------------------------------------
extern "C" void kernel_launch(void* const* d_in, const int* in_sizes, int n_in,
                              void* d_out, int out_size, void* d_ws, size_t ws_size,
                              hipStream_t stream)
{
    const void*  cache_k = d_in[0];
    const void*  cache_v = d_in[1];
    const f32x4* new_k   = (const f32x4*)d_in[2];
    const f32x4* new_v   = (const f32x4*)d_in[3];
    const int*   pos     = (const int*)  d_in[4];

    // Bulk copy: 256 MiB per tensor / 32 KiB per WG = 8192 blocks; y = {k,v}.
    {
        dim3 grid(BYTES_PER_TENSOR / BLOCK_CHUNK, 2);
        cache_copy_async_kernel<<<grid, 256, 0, stream>>>(cache_k, cache_v, d_out);
    }

    // Scatter: 2,097,152 float4s, one per thread -> 8192 blocks.
    {
        const int threads = 256;
        const int blocks  = (int)((2 * NNEW4 + threads - 1) / threads);  // 8192
        cache_scatter_kernel<<<blocks, threads, 0, stream>>>(new_k, new_v, pos,
                                                             (f32x4*)d_out);
    }
}